// HighChebNet_10179072492117
// MI455X (gfx1250) — compile-verified
//
#include <hip/hip_runtime.h>
#include <hip/hip_bf16.h>

// ---------------------------------------------------------------------------
// Problem dims (from setup_inputs): B=32, IU=128, OU=128, ID=64, VD=64
// Key algebraic simplification: votes.mean(axis=1) is a single GEMM
//   mean[b, o*64+d] = (1/128) * sum_k W2d[k, o*64+d] * x2d[b, k],  k=(i,e), K=8192
// That GEMM streams the 256 MB W_vote exactly once -> memory bound at 23.3 TB/s.
// ---------------------------------------------------------------------------

typedef float v2f  __attribute__((ext_vector_type(2)));
typedef float v8f  __attribute__((ext_vector_type(8)));

#define B_    32
#define IU_   128
#define OU_   128
#define ID_   64
#define VD_   64
#define KBIG  (IU_ * ID_)   // 8192
#define NBIG  (OU_ * VD_)   // 8192
#define L_    128           // attention sequence length (= IU)
#define D_    64            // head dim (= VD = ID)
#define WSTR  (OU_ * VD_ * ID_)   // W_vote stride over i: 524288 floats

static __device__ __forceinline__ v8f wmma_f32_4(v2f a, v2f b, v8f c) {
  // V_WMMA_F32_16X16X4_F32 : D(16x16,f32) = A(16x4,f32) * B(4x16,f32) + C
  return __builtin_amdgcn_wmma_f32_16x16x4_f32(
      /*neg_a=*/false, a, /*neg_b=*/false, b,
      /*c_mod=*/(short)0, c, /*reuse_a=*/false, /*reuse_b=*/false);
}

// ---------------------------------------------------------------------------
// Kernel 1: mean-of-votes GEMM.  M=32 (batch), N=8192, K=8192.
// 4 waves per block; each wave owns a 2048-wide K slice of the same 16-wide N
// tile (2048 waves total -> enough independent streams for 23.3 TB/s).
// Uniform outer loop over i-blocks (prefetch once per block, no divergence);
// fully unrolled inner loop over e uses immediate-offset b64 loads.
// Partials reduced through LDS.
// ---------------------------------------------------------------------------
__global__ __launch_bounds__(128)
void votes_mean_gemm(const float* __restrict__ X,   // (32, 8192) row-major
                     const float* __restrict__ W,   // (128,128,64,64)
                     float* __restrict__ Y)         // (32, 8192)
{
  __shared__ float red[4][32][16];   // 8 KB partial sums

  const int wave = threadIdx.x >> 5;
  const int lane = threadIdx.x & 31;
  const int lr   = lane & 15;
  const int half = lane >> 4;

  const int n0 = blockIdx.x * 16;
  const int n  = n0 + lr;            // this lane's output column
  const int o  = n >> 6;
  const int d  = n & 63;
  const float* wcol = W + (long long)o * (VD_ * ID_) + (long long)d * ID_
                        + 2 * half;

  const int ibeg = wave * (IU_ / 4);       // 32 i-blocks per wave

  v8f c0 = {};
  v8f c1 = {};

  for (int ib = 0; ib < IU_ / 4; ++ib) {
    const int i = ibeg + ib;
    const float* wp  = wcol + (long long)i * WSTR;
    __builtin_prefetch(wp + WSTR, 0, 3);              // next i-block, uniform
    const float* xp0 = X + (long long)lr        * KBIG + i * ID_ + 2 * half;
    const float* xp1 = X + (long long)(16 + lr) * KBIG + i * ID_ + 2 * half;
    #pragma unroll
    for (int kk = 0; kk < ID_; kk += 4) {
      v2f bf = *(const v2f*)(wp  + kk);
      v2f a0 = *(const v2f*)(xp0 + kk);
      v2f a1 = *(const v2f*)(xp1 + kk);
      c0 = wmma_f32_4(a0, bf, c0);
      c1 = wmma_f32_4(a1, bf, c1);
    }
  }

  #pragma unroll
  for (int j = 0; j < 8; ++j) {
    red[wave][lane][j]     = c0[j];
    red[wave][lane][8 + j] = c1[j];
  }
  __syncthreads();

  if (wave == 0) {
    const float inv_iu = 1.0f / (float)IU_;
    #pragma unroll
    for (int j = 0; j < 8; ++j) {
      float s0 = red[0][lane][j]     + red[1][lane][j]
               + red[2][lane][j]     + red[3][lane][j];
      float s1 = red[0][lane][8 + j] + red[1][lane][8 + j]
               + red[2][lane][8 + j] + red[3][lane][8 + j];
      const int r0 = j + 8 * half;          // rows 0..15 of this C tile
      Y[(long long)r0 * NBIG + n]        = s0 * inv_iu;
      Y[(long long)(16 + r0) * NBIG + n] = s1 * inv_iu;
    }
  }
}

// ---------------------------------------------------------------------------
// Kernel 2: LayerNorm over last dim (64), rows = B*OU = 4096. Wave per row.
// ---------------------------------------------------------------------------
__global__ __launch_bounds__(256)
void layernorm64(const float* __restrict__ in, const float* __restrict__ g,
                 const float* __restrict__ bta, float* __restrict__ out)
{
  const int wave = threadIdx.x >> 5;
  const int lane = threadIdx.x & 31;
  const int row  = blockIdx.x * 8 + wave;

  v2f xv = *(const v2f*)(in + (long long)row * 64 + lane * 2);
  float s  = xv.x + xv.y;
  float s2 = xv.x * xv.x + xv.y * xv.y;
  #pragma unroll
  for (int m = 1; m <= 16; m <<= 1) {
    s  += __shfl_xor(s,  m, 32);
    s2 += __shfl_xor(s2, m, 32);
  }
  const float mu   = s * (1.0f / 64.0f);
  const float var  = s2 * (1.0f / 64.0f) - mu * mu;
  const float rstd = rsqrtf(var + 1e-6f);

  v2f gv = *(const v2f*)(g + lane * 2);
  v2f bv = *(const v2f*)(bta + lane * 2);
  v2f yv;
  yv.x = (xv.x - mu) * rstd * gv.x + bv.x;
  yv.y = (xv.y - mu) * rstd * gv.y + bv.y;
  *(v2f*)(out + (long long)row * 64 + lane * 2) = yv;
}

// ---------------------------------------------------------------------------
// Kernel 3: generic GEMM with N=64:  C[r, 0..63] = A[r, :K] @ Bm[K, 64] (+bias)
// K is a template parameter (64 or 128) so the reduction loop fully unrolls.
// grid.x = row tiles (16 rows each), grid.y = batch. One wave per block.
// ---------------------------------------------------------------------------
template<int K>
__global__ __launch_bounds__(32)
void gemm_n64(const float* __restrict__ A, const float* __restrict__ Bm,
              const float* __restrict__ bias, float* __restrict__ C,
              long long sA, long long sB, long long sC)
{
  A  += (long long)blockIdx.y * sA;
  Bm += (long long)blockIdx.y * sB;
  C  += (long long)blockIdx.y * sC;

  const int lane = threadIdx.x & 31;
  const int lr   = lane & 15;
  const int half = lane >> 4;
  const int row0 = blockIdx.x * 16;

  v8f c[4] = {};

  const float* ap = A + (long long)(row0 + lr) * K + 2 * half;

  #pragma unroll
  for (int k = 0; k < K; k += 4) {
    v2f a = *(const v2f*)(ap + k);
    #pragma unroll
    for (int t = 0; t < 4; ++t) {
      const int col = t * 16 + lr;
      v2f bf;
      bf.x = Bm[(k + 2 * half)     * 64 + col];
      bf.y = Bm[(k + 2 * half + 1) * 64 + col];
      c[t] = wmma_f32_4(a, bf, c[t]);
    }
  }

  #pragma unroll
  for (int t = 0; t < 4; ++t) {
    const int col = t * 16 + lr;
    const float bv = bias ? bias[col] : 0.0f;
    #pragma unroll
    for (int j = 0; j < 8; ++j) {
      const int r = row0 + j + 8 * half;
      C[(long long)r * 64 + col] = c[t][j] + bv;
    }
  }
}

// ---------------------------------------------------------------------------
// Kernel 4: attention scores.  One block (8 waves) per batch element.
// S[b] = softmax(Q[b] @ K[b]^T * scale), Q,K: (128,64), S: (128,128).
// K matrix staged into LDS with CDNA5 async global->LDS copies (ASYNCcnt),
// bypassing VGPRs entirely; transposed B fragments are contiguous float2
// ds_load_b64 reads. Softmax in registers + 16-lane butterflies.
// ---------------------------------------------------------------------------
__global__ __launch_bounds__(256)
void attn_scores(const float* __restrict__ Q, const float* __restrict__ Km,
                 float* __restrict__ S, float scale)
{
  __shared__ __align__(16) float Kl[L_ * D_];   // 32 KB

  const int b = blockIdx.x;
  Q  += (long long)b * L_ * D_;
  Km += (long long)b * L_ * D_;
  S  += (long long)b * L_ * L_;

  // Async-stage K into LDS: global_load_async_to_lds_b128 writes LDS directly
  // (no VGPR round-trip). Low 32 bits of a generic pointer to __shared__ are
  // the LDS byte offset (ISA 10.2 aperture address calc).
  #pragma unroll
  for (int it = 0; it < 8; ++it) {
    const int i        = it * 256 + threadIdx.x;
    unsigned lds_off   = (unsigned)(uintptr_t)(&Kl[i * 4]);
    const float4* gsrc = ((const float4*)Km) + i;
    asm volatile("global_load_async_to_lds_b128 %0, %1, off"
                 :: "v"(lds_off), "v"(gsrc) : "memory");
  }
  asm volatile("s_wait_asynccnt 0" ::: "memory");
  __syncthreads();

  const int wave = threadIdx.x >> 5;
  const int lane = threadIdx.x & 31;
  const int lr   = lane & 15;
  const int half = lane >> 4;
  const int q0   = wave * 16;          // 16 query rows per wave, 8 waves = 128

  v8f acc[8] = {};

  const float* qp = Q + (long long)(q0 + lr) * D_ + 2 * half;

  #pragma unroll
  for (int k = 0; k < D_; k += 4) {
    v2f a = *(const v2f*)(qp + k);
    #pragma unroll
    for (int nt = 0; nt < 8; ++nt) {
      const int j = nt * 16 + lr;                         // key index (B column)
      v2f bf = *(const v2f*)(&Kl[j * D_ + k + 2 * half]); // B[e][j] = K[j][e]
      acc[nt] = wmma_f32_4(a, bf, acc[nt]);
    }
  }

  // softmax over each row (128 values = 8 tiles x 16 lanes within a half)
  #pragma unroll
  for (int j = 0; j < 8; ++j) {
    float m = -3.4e38f;
    #pragma unroll
    for (int nt = 0; nt < 8; ++nt) m = fmaxf(m, acc[nt][j] * scale);
    #pragma unroll
    for (int msk = 1; msk <= 8; msk <<= 1) m = fmaxf(m, __shfl_xor(m, msk, 32));

    float s = 0.0f;
    #pragma unroll
    for (int nt = 0; nt < 8; ++nt) {
      float e = __expf(acc[nt][j] * scale - m);
      acc[nt][j] = e;
      s += e;
    }
    #pragma unroll
    for (int msk = 1; msk <= 8; msk <<= 1) s += __shfl_xor(s, msk, 32);
    const float inv = 1.0f / s;

    const int row = q0 + j + 8 * half;
    #pragma unroll
    for (int nt = 0; nt < 8; ++nt)
      S[(long long)row * L_ + nt * 16 + lr] = acc[nt][j] * inv;
  }
}

// ---------------------------------------------------------------------------
// Host launch
// ---------------------------------------------------------------------------
extern "C" void kernel_launch(void* const* d_in, const int* in_sizes, int n_in,
                              void* d_out, int out_size, void* d_ws, size_t ws_size,
                              hipStream_t stream) {
  const float* x    = (const float*)d_in[0];
  const float* Wv   = (const float*)d_in[1];
  const float* ln_g = (const float*)d_in[2];
  const float* ln_b = (const float*)d_in[3];
  const float* wq1 = (const float*)d_in[4];  const float* bq1 = (const float*)d_in[5];
  const float* wk1 = (const float*)d_in[6];  const float* bk1 = (const float*)d_in[7];
  const float* wv1 = (const float*)d_in[8];  const float* bv1 = (const float*)d_in[9];
  const float* wo1 = (const float*)d_in[10]; const float* bo1 = (const float*)d_in[11];
  const float* wq2 = (const float*)d_in[12]; const float* bq2 = (const float*)d_in[13];
  const float* wk2 = (const float*)d_in[14]; const float* bk2 = (const float*)d_in[15];
  const float* wv2 = (const float*)d_in[16]; const float* bv2 = (const float*)d_in[17];
  const float* wo2 = (const float*)d_in[18]; const float* bo2 = (const float*)d_in[19];

  float* out = (float*)d_out;
  float* diffused = out + 0;                     // (32,128,64)
  float* agg      = out + 262144;                // (32,128,64)
  float* fwd      = out + 524288;                // (32,128,128)
  float* bwd      = out + 1048576;               // (32,128,128)

  float* ws    = (float*)d_ws;
  float* meanv = ws + 0;                         // (32, 8192)
  float* normv = ws + 262144;                    // (32,128,64) viewed as (32,8192)
  float* Qb    = ws + 524288;                    // (32,128,64)
  float* Kb    = ws + 786432;                    // (32,128,64)
  float* Vb    = ws + 1048576;                   // (32,128,64)
  float* Ob    = ws + 1310720;                   // (32,128,64)

  const long long sQ = (long long)L_ * D_;       // 8192
  const long long sS = (long long)L_ * L_;       // 16384

  // 1) mean of capsule votes (the 256 MB streaming GEMM), K split over 4 waves
  votes_mean_gemm<<<NBIG / 16, 128, 0, stream>>>(x, Wv, meanv);

  // 2) layer norm -> norm_value
  layernorm64<<<(B_ * OU_) / 8, 256, 0, stream>>>(meanv, ln_g, ln_b, normv);

  // 3) attention 1 projections: Q=x@wq1+bq1, K=norm@wk1+bk1, V=x@wv1+bv1
  gemm_n64<64><<<dim3(256, 1), 32, 0, stream>>>(x,     wq1, bq1, Qb, 0, 0, 0);
  gemm_n64<64><<<dim3(256, 1), 32, 0, stream>>>(normv, wk1, bk1, Kb, 0, 0, 0);
  gemm_n64<64><<<dim3(256, 1), 32, 0, stream>>>(x,     wv1, bv1, Vb, 0, 0, 0);

  // 4) fwd_attn = softmax(Q K^T / 8)   (written directly to output region)
  attn_scores<<<B_, 256, 0, stream>>>(Qb, Kb, fwd, 0.125f);

  // 5) O = S @ V  (batched), then aggregated = O @ wo1 + bo1
  gemm_n64<128><<<dim3(8, B_), 32, 0, stream>>>(fwd, Vb, nullptr, Ob, sS, sQ, sQ);
  gemm_n64<64><<<dim3(256, 1), 32, 0, stream>>>(Ob, wo1, bo1, agg, 0, 0, 0);

  // 6) attention 2 projections: Q=agg@wq2+bq2, K=x@wk2+bk2, V=agg@wv2+bv2
  gemm_n64<64><<<dim3(256, 1), 32, 0, stream>>>(agg, wq2, bq2, Qb, 0, 0, 0);
  gemm_n64<64><<<dim3(256, 1), 32, 0, stream>>>(x,   wk2, bk2, Kb, 0, 0, 0);
  gemm_n64<64><<<dim3(256, 1), 32, 0, stream>>>(agg, wv2, bv2, Vb, 0, 0, 0);

  // 7) bwd_attn = softmax(Q2 K2^T / 8)
  attn_scores<<<B_, 256, 0, stream>>>(Qb, Kb, bwd, 0.125f);

  // 8) O2 = S2 @ V2, diffused = O2 @ wo2 + bo2
  gemm_n64<128><<<dim3(8, B_), 32, 0, stream>>>(bwd, Vb, nullptr, Ob, sS, sQ, sQ);
  gemm_n64<64><<<dim3(256, 1), 32, 0, stream>>>(Ob, wo2, bo2, diffused, 0, 0, 0);
}